// Geoattn_GNN_25890062860777
// MI455X (gfx1250) — compile-verified
//
#include <hip/hip_runtime.h>
#include <math.h>

#define SLOPE 0.2f

typedef float v2f __attribute__((ext_vector_type(2)));
typedef float v8f __attribute__((ext_vector_type(8)));

__device__ __forceinline__ v8f wmma4(v2f a, v2f b, v8f c) {
  // D = A(16x4) * B(4x16) + C, all f32  ->  v_wmma_f32_16x16x4_f32
  return __builtin_amdgcn_wmma_f32_16x16x4_f32(false, a, false, b, (short)0, c,
                                               false, false);
}

__device__ __forceinline__ void atomicMaxF(float* addr, float v) {
  if (v >= 0.0f) atomicMax((int*)addr, __float_as_int(v));
  else           atomicMin((unsigned int*)addr, __float_as_uint(v));
}

// ---------------- K0: init accumulators ----------------
__global__ void k_init(float* seg_max, float* seg_sum, float* emb_sca,
                       float* emb_vec, int N) {
  int i = blockIdx.x * blockDim.x + threadIdx.x;
  if (i < N * 16) { seg_max[i] = -INFINITY; seg_sum[i] = 0.f; emb_sca[i] = 0.f; }
  if (i < N * 48) emb_vec[i] = 0.f;
}

// ---------------- K1a: node GEMMs (13 -> 16, two weights, shared A) ----------
__global__ __launch_bounds__(256)
void k_node_gemm(const float* __restrict__ node_sca,
                 const float* __restrict__ W_node, const float* __restrict__ b_node,
                 const float* __restrict__ W_nsca, const float* __restrict__ b_nsca,
                 float* __restrict__ node_lin, float* __restrict__ node_hid, int N) {
  int lane = threadIdx.x & 31, wave = threadIdx.x >> 5;
  int nb = (blockIdx.x * 8 + wave) * 16;
  if (nb >= N) return;
  int half = lane >> 4, m = lane & 15, o = lane & 15;
  int na = nb + m; if (na >= N) na = N - 1;
  v8f c0 = {}, c1 = {};
  #pragma unroll
  for (int kk = 0; kk < 16; kk += 4) {
    int k0 = kk + half * 2;
    v2f a, b0, b1;
    a.x  = (k0     < 13) ? node_sca[na * 13 + k0]     : 0.f;
    a.y  = (k0 + 1 < 13) ? node_sca[na * 13 + k0 + 1] : 0.f;
    b0.x = (k0     < 13) ? W_node[o * 13 + k0]        : 0.f;
    b0.y = (k0 + 1 < 13) ? W_node[o * 13 + k0 + 1]    : 0.f;
    b1.x = (k0     < 13) ? W_nsca[o * 13 + k0]        : 0.f;
    b1.y = (k0 + 1 < 13) ? W_nsca[o * 13 + k0 + 1]    : 0.f;
    c0 = wmma4(a, b0, c0);
    c1 = wmma4(a, b1, c1);
  }
  float bb0 = b_node[o], bb1 = b_nsca[o];
  #pragma unroll
  for (int r = 0; r < 8; r++) {
    int node = nb + r + half * 8;
    if (node < N) {
      node_lin[node * 16 + o] = c0[r] + bb0;
      node_hid[node * 16 + o] = c1[r] + bb1;
    }
  }
}

// ---------------- K1b: node VN linears (K=2, elementwise) --------------------
__global__ void k_node_vn(const float* __restrict__ node_vec,
                          const float* __restrict__ Wva, const float* __restrict__ bva,
                          const float* __restrict__ Wnv,
                          float* __restrict__ avh, float* __restrict__ nvh, int N) {
  int t = blockIdx.x * blockDim.x + threadIdx.x;
  if (t >= N * 16) return;
  int n = t >> 4, o = t & 15;
  float w0 = Wva[o * 2], w1 = Wva[o * 2 + 1], b = bva[o];
  float u0 = Wnv[o * 2], u1 = Wnv[o * 2 + 1];
  #pragma unroll
  for (int d = 0; d < 3; d++) {
    float x0 = node_vec[n * 6 + d];       // [n][0][d]
    float x1 = node_vec[n * 6 + 3 + d];   // [n][1][d]
    avh[n * 48 + o * 3 + d] = x0 * w0 + x1 * w1 + b;
    nvh[n * 48 + o * 3 + d] = x0 * u0 + x1 * u1;
  }
}

// ----- shared: 16-edge tile of attention logits (27->16 GEMM via 7 WMMAs) ----
__device__ __forceinline__ v8f edge_logit_tile(
    const float* __restrict__ node_sca, const float* __restrict__ node_pos,
    const int* __restrict__ srcI, const int* __restrict__ dstI,
    const float* __restrict__ Wsa, const float* __restrict__ bsa,
    int ebase, int E, int lane) {
  int half = lane >> 4, m = lane & 15, o = lane & 15;
  int e = ebase + m; if (e >= E) e = E - 1;
  int s = srcI[e], dd = dstI[e];
  float dx = node_pos[s * 3 + 0] - node_pos[dd * 3 + 0];
  float dy = node_pos[s * 3 + 1] - node_pos[dd * 3 + 1];
  float dz = node_pos[s * 3 + 2] - node_pos[dd * 3 + 2];
  float dist = sqrtf(dx * dx + dy * dy + dz * dz);
  auto feat = [&](int k) -> float {
    if (k < 13) return node_sca[s * 13 + k];
    if (k < 26) return node_sca[dd * 13 + (k - 13)];
    if (k == 26) return dist;
    return 0.f;
  };
  v8f c = {};
  #pragma unroll
  for (int kk = 0; kk < 28; kk += 4) {
    int k0 = kk + half * 2;
    v2f a, b;
    a.x = feat(k0);
    a.y = feat(k0 + 1);
    b.x = (k0     < 27) ? Wsa[o * 27 + k0]     : 0.f;
    b.y = (k0 + 1 < 27) ? Wsa[o * 27 + k0 + 1] : 0.f;
    c = wmma4(a, b, c);
  }
  float bo = bsa[o];
  #pragma unroll
  for (int r = 0; r < 8; r++) c[r] += bo;
  return c;
}

// ---------------- K2: segment max of logits over src -------------------------
__global__ __launch_bounds__(256)
void k_edge_max(const float* __restrict__ node_sca, const float* __restrict__ node_pos,
                const int* __restrict__ srcI, const int* __restrict__ dstI,
                const float* __restrict__ Wsa, const float* __restrict__ bsa,
                float* __restrict__ seg_max, int E) {
  int lane = threadIdx.x & 31, wave = threadIdx.x >> 5;
  int ebase = (blockIdx.x * 8 + wave) * 16;
  if (ebase >= E) return;
  v8f c = edge_logit_tile(node_sca, node_pos, srcI, dstI, Wsa, bsa, ebase, E, lane);
  int half = lane >> 4, o = lane & 15;
  #pragma unroll
  for (int r = 0; r < 8; r++) {
    int e = ebase + r + half * 8;
    if (e < E) atomicMaxF(&seg_max[srcI[e] * 16 + o], c[r]);
  }
}

// ---------------- K3a: segment sum of exp(logit - max) -----------------------
__global__ __launch_bounds__(256)
void k_edge_expsum(const float* __restrict__ node_sca, const float* __restrict__ node_pos,
                   const int* __restrict__ srcI, const int* __restrict__ dstI,
                   const float* __restrict__ Wsa, const float* __restrict__ bsa,
                   const float* __restrict__ seg_max, float* __restrict__ seg_sum, int E) {
  int lane = threadIdx.x & 31, wave = threadIdx.x >> 5;
  int ebase = (blockIdx.x * 8 + wave) * 16;
  if (ebase >= E) return;
  v8f c = edge_logit_tile(node_sca, node_pos, srcI, dstI, Wsa, bsa, ebase, E, lane);
  int half = lane >> 4, o = lane & 15;
  #pragma unroll
  for (int r = 0; r < 8; r++) {
    int e = ebase + r + half * 8;
    if (e < E) {
      int s = srcI[e];
      atomicAdd(&seg_sum[s * 16 + o], expf(c[r] - seg_max[s * 16 + o]));
    }
  }
}

// ---------------- K3b: vector attention (48-dim dot, per edge) ---------------
__global__ void k_alpha_vec(const int* __restrict__ srcI, const int* __restrict__ dstI,
                            const float* __restrict__ avh, float* __restrict__ alpha_vec,
                            int E) {
  int e = blockIdx.x * blockDim.x + threadIdx.x;
  if (e >= E) return;
  const float* pa = avh + srcI[e] * 48;
  const float* pb = avh + dstI[e] * 48;
  float acc = 0.f;
  #pragma unroll 8
  for (int i = 0; i < 48; i++) acc += pa[i] * pb[i];
  alpha_vec[e] = 1.f / (1.f + expf(-acc));
}

// ---------------- K4: messages + segment sums --------------------------------
__global__ __launch_bounds__(256)
void k_edge_msg(const float* __restrict__ node_sca, const float* __restrict__ node_pos,
                const int* __restrict__ srcI, const int* __restrict__ dstI,
                const float* __restrict__ Wsa, const float* __restrict__ bsa,
                const float* __restrict__ edge_sca, const float* __restrict__ edge_vec,
                const float* __restrict__ W_edge, const float* __restrict__ b_edge,
                const float* __restrict__ W_esca, const float* __restrict__ b_esca,
                const float* __restrict__ W_evec,
                const float* __restrict__ node_lin, const float* __restrict__ node_hid,
                const float* __restrict__ nvh,
                const float* __restrict__ seg_max, const float* __restrict__ seg_sum,
                const float* __restrict__ alpha_vec,
                float* __restrict__ emb_sca, float* __restrict__ emb_vec, int E) {
  int lane = threadIdx.x & 31, wave = threadIdx.x >> 5;
  int ebase = (blockIdx.x * 8 + wave) * 16;
  if (ebase >= E) return;
  v8f clog = edge_logit_tile(node_sca, node_pos, srcI, dstI, Wsa, bsa, ebase, E, lane);
  int half = lane >> 4, m = lane & 15, o = lane & 15;
  // edge 4->16 GEMMs (single K=4 WMMA each, shared A)
  int ea = ebase + m; if (ea >= E) ea = E - 1;
  int k0 = half * 2;
  v2f a, b0, b1;
  a.x  = edge_sca[ea * 4 + k0];     a.y  = edge_sca[ea * 4 + k0 + 1];
  b0.x = W_edge[o * 4 + k0];        b0.y = W_edge[o * 4 + k0 + 1];
  b1.x = W_esca[o * 4 + k0];        b1.y = W_esca[o * 4 + k0 + 1];
  v8f cel = {}, ceh = {};
  cel = wmma4(a, b0, cel);
  ceh = wmma4(a, b1, ceh);
  float be = b_edge[o], bes = b_esca[o], wev = W_evec[o];
  #pragma unroll
  for (int r = 0; r < 8; r++) {
    int e = ebase + r + half * 8;
    if (e < E) {
      int s = srcI[e];
      float asca = expf(clog[r] - seg_max[s * 16 + o]) / seg_sum[s * 16 + o];
      float el   = cel[r] + be;
      atomicAdd(&emb_sca[s * 16 + o], node_lin[s * 16 + o] * el * asca);
      float nh  = node_hid[s * 16 + o];
      float esh = ceh[r] + bes;
      float av  = alpha_vec[e];
      #pragma unroll
      for (int d = 0; d < 3; d++) {
        float ev = edge_vec[e * 3 + d] * wev;
        float nv = nvh[s * 48 + o * 3 + d];
        atomicAdd(&emb_vec[s * 48 + o * 3 + d], (nh * ev + nv * esh) * av);
      }
    }
  }
}

// ---------------- K5: GVPerceptron + VN nonlinearities (per 16-node tile) ----
__device__ __forceinline__ void tile_store(float* T, v8f c, int lane) {
  int half = lane >> 4, col = lane & 15;
  #pragma unroll
  for (int r = 0; r < 8; r++) T[(r + half * 8) * 17 + col] = c[r];
}
__device__ __forceinline__ v2f tile_frag(const float* T, int lane, int kk) {
  int half = lane >> 4, m = lane & 15, k0 = kk + half * 2;
  v2f a; a.x = T[m * 17 + k0]; a.y = T[m * 17 + k0 + 1]; return a;
}

__global__ __launch_bounds__(256)
void k_node_out(const float* __restrict__ emb_sca, const float* __restrict__ emb_vec,
                const float* __restrict__ Wv1, const float* __restrict__ Wv2,
                const float* __restrict__ Wgs,
                const float* __restrict__ Wg, const float* __restrict__ bg,
                const float* __restrict__ Wd,
                float* __restrict__ out_sca, float* __restrict__ out_vec, int N) {
  __shared__ float lds[8 * 16 * 17];
  int lane = threadIdx.x & 31, wave = threadIdx.x >> 5;
  int nb = (blockIdx.x * 8 + wave) * 16;
  if (nb >= N) return;
  float* T = lds + wave * (16 * 17);
  int half = lane >> 4, m = lane & 15, o = lane & 15;
  int na = nb + m; if (na >= N) na = N - 1;

  // v1 = VN(emb_vec, Wv1) : 3 x (K=16) GEMMs
  v8f cv1[3];
  #pragma unroll
  for (int d = 0; d < 3; d++) {
    v8f c = {};
    #pragma unroll
    for (int kk = 0; kk < 16; kk += 4) {
      int k0 = kk + half * 2;
      v2f aa, bb;
      aa.x = emb_vec[na * 48 + k0 * 3 + d];
      aa.y = emb_vec[na * 48 + (k0 + 1) * 3 + d];
      bb.x = Wv1[o * 16 + k0]; bb.y = Wv1[o * 16 + k0 + 1];
      c = wmma4(aa, bb, c);
    }
    cv1[d] = c;
  }
  // vnorm
  v8f vn;
  #pragma unroll
  for (int r = 0; r < 8; r++)
    vn[r] = sqrtf(cv1[0][r] * cv1[0][r] + cv1[1][r] * cv1[1][r] +
                  cv1[2][r] * cv1[2][r] + 1e-12f);

  // out_sca = [vnorm, emb_sca] @ Wgs.T   (K=32)
  tile_store(T, vn, lane); __threadfence_block();
  v8f cos_ = {};
  #pragma unroll
  for (int kk = 0; kk < 16; kk += 4) {
    v2f aa = tile_frag(T, lane, kk);
    int k0 = kk + half * 2;
    v2f bb; bb.x = Wgs[o * 32 + k0]; bb.y = Wgs[o * 32 + k0 + 1];
    cos_ = wmma4(aa, bb, cos_);
  }
  __threadfence_block();
  #pragma unroll
  for (int kk = 0; kk < 16; kk += 4) {
    int k0 = kk + half * 2;
    v2f aa, bb;
    aa.x = emb_sca[na * 16 + k0]; aa.y = emb_sca[na * 16 + k0 + 1];
    bb.x = Wgs[o * 32 + 16 + k0]; bb.y = Wgs[o * 32 + 16 + k0 + 1];
    cos_ = wmma4(aa, bb, cos_);
  }

  // gate = sigmoid(out_sca @ Wg.T + bg)
  tile_store(T, cos_, lane); __threadfence_block();
  v8f cg = {};
  #pragma unroll
  for (int kk = 0; kk < 16; kk += 4) {
    v2f aa = tile_frag(T, lane, kk);
    int k0 = kk + half * 2;
    v2f bb; bb.x = Wg[o * 16 + k0]; bb.y = Wg[o * 16 + k0 + 1];
    cg = wmma4(aa, bb, cg);
  }
  __threadfence_block();
  float bgo = bg[o];
  v8f g;
  #pragma unroll
  for (int r = 0; r < 8; r++) g[r] = 1.f / (1.f + expf(-(cg[r] + bgo)));

  // out_vec = gate * VN(v1, Wv2)
  v8f ov[3];
  #pragma unroll
  for (int d = 0; d < 3; d++) {
    tile_store(T, cv1[d], lane); __threadfence_block();
    v8f c = {};
    #pragma unroll
    for (int kk = 0; kk < 16; kk += 4) {
      v2f aa = tile_frag(T, lane, kk);
      int k0 = kk + half * 2;
      v2f bb; bb.x = Wv2[o * 16 + k0]; bb.y = Wv2[o * 16 + k0 + 1];
      c = wmma4(aa, bb, c);
    }
    __threadfence_block();
    #pragma unroll
    for (int r = 0; r < 8; r++) ov[d][r] = g[r] * c[r];
  }

  // leaky relu on out_sca, store
  #pragma unroll
  for (int r = 0; r < 8; r++) {
    int node = nb + r + half * 8;
    float x = cos_[r];
    if (node < N) out_sca[node * 16 + o] = (x >= 0.f) ? x : SLOPE * x;
  }

  // dir = VN(out_vec, Wd)
  v8f dir[3];
  #pragma unroll
  for (int d = 0; d < 3; d++) {
    tile_store(T, ov[d], lane); __threadfence_block();
    v8f c = {};
    #pragma unroll
    for (int kk = 0; kk < 16; kk += 4) {
      v2f aa = tile_frag(T, lane, kk);
      int k0 = kk + half * 2;
      v2f bb; bb.x = Wd[o * 16 + k0]; bb.y = Wd[o * 16 + k0 + 1];
      c = wmma4(aa, bb, c);
    }
    __threadfence_block();
    dir[d] = c;
  }

  // VN leaky relu projection, store
  #pragma unroll
  for (int r = 0; r < 8; r++) {
    float dot = ov[0][r] * dir[0][r] + ov[1][r] * dir[1][r] + ov[2][r] * dir[2][r];
    float den = dir[0][r] * dir[0][r] + dir[1][r] * dir[1][r] +
                dir[2][r] * dir[2][r] + 1e-6f;
    float coef = dot / den;
    int node = nb + r + half * 8;
    #pragma unroll
    for (int d = 0; d < 3; d++) {
      float v = ov[d][r];
      float proj = v - coef * dir[d][r];
      float sel = (dot >= 0.f) ? v : proj;
      if (node < N) out_vec[node * 48 + o * 3 + d] = SLOPE * v + (1.f - SLOPE) * sel;
    }
  }
}

extern "C" void kernel_launch(void* const* d_in, const int* in_sizes, int n_in,
                              void* d_out, int out_size, void* d_ws, size_t ws_size,
                              hipStream_t stream) {
  const float* node_sca  = (const float*)d_in[0];
  const float* node_vec  = (const float*)d_in[1];
  const float* node_pos  = (const float*)d_in[2];
  const float* edge_sca  = (const float*)d_in[3];
  const float* edge_vec  = (const float*)d_in[4];
  const int*   edge_index = (const int*)d_in[5];
  const float* W_sca_attn = (const float*)d_in[6];  const float* b_sca_attn = (const float*)d_in[7];
  const float* W_vec_attn = (const float*)d_in[8];  const float* b_vec_attn = (const float*)d_in[9];
  const float* W_node     = (const float*)d_in[10]; const float* b_node     = (const float*)d_in[11];
  const float* W_edge     = (const float*)d_in[12]; const float* b_edge     = (const float*)d_in[13];
  const float* W_node_sca = (const float*)d_in[14]; const float* b_node_sca = (const float*)d_in[15];
  const float* W_node_vec = (const float*)d_in[16];
  const float* W_edge_sca = (const float*)d_in[17]; const float* b_edge_sca = (const float*)d_in[18];
  const float* W_edge_vec = (const float*)d_in[19];
  const float* W_gv_vec1  = (const float*)d_in[20];
  const float* W_gv_vec2  = (const float*)d_in[21];
  const float* W_gv_sca   = (const float*)d_in[22];
  const float* W_gate     = (const float*)d_in[23]; const float* b_gate     = (const float*)d_in[24];
  const float* W_dir      = (const float*)d_in[25];

  int N = in_sizes[0] / 13;
  int E = in_sizes[3] / 4;
  const int* srcI = edge_index;
  const int* dstI = edge_index + E;

  float* ws = (float*)d_ws;
  float* node_lin = ws;
  float* node_hid = node_lin + (size_t)16 * N;
  float* nvh      = node_hid + (size_t)16 * N;
  float* avh      = nvh      + (size_t)48 * N;
  float* seg_max  = avh      + (size_t)48 * N;
  float* seg_sum  = seg_max  + (size_t)16 * N;
  float* emb_sca  = seg_sum  + (size_t)16 * N;
  float* emb_vec  = emb_sca  + (size_t)16 * N;
  float* alpha_v  = emb_vec  + (size_t)48 * N;

  float* out_sca = (float*)d_out;
  float* out_vec = out_sca + (size_t)16 * N;

  int ntiles = (N + 15) / 16, etiles = (E + 15) / 16;

  k_init<<<(48 * N + 255) / 256, 256, 0, stream>>>(seg_max, seg_sum, emb_sca, emb_vec, N);
  k_node_gemm<<<(ntiles + 7) / 8, 256, 0, stream>>>(node_sca, W_node, b_node,
                                                    W_node_sca, b_node_sca,
                                                    node_lin, node_hid, N);
  k_node_vn<<<(N * 16 + 255) / 256, 256, 0, stream>>>(node_vec, W_vec_attn, b_vec_attn,
                                                      W_node_vec, avh, nvh, N);
  k_edge_max<<<(etiles + 7) / 8, 256, 0, stream>>>(node_sca, node_pos, srcI, dstI,
                                                   W_sca_attn, b_sca_attn, seg_max, E);
  k_edge_expsum<<<(etiles + 7) / 8, 256, 0, stream>>>(node_sca, node_pos, srcI, dstI,
                                                      W_sca_attn, b_sca_attn,
                                                      seg_max, seg_sum, E);
  k_alpha_vec<<<(E + 255) / 256, 256, 0, stream>>>(srcI, dstI, avh, alpha_v, E);
  k_edge_msg<<<(etiles + 7) / 8, 256, 0, stream>>>(node_sca, node_pos, srcI, dstI,
                                                   W_sca_attn, b_sca_attn,
                                                   edge_sca, edge_vec,
                                                   W_edge, b_edge, W_edge_sca, b_edge_sca,
                                                   W_edge_vec,
                                                   node_lin, node_hid, nvh,
                                                   seg_max, seg_sum, alpha_v,
                                                   emb_sca, emb_vec, E);
  k_node_out<<<(ntiles + 7) / 8, 256, 0, stream>>>(emb_sca, emb_vec,
                                                   W_gv_vec1, W_gv_vec2, W_gv_sca,
                                                   W_gate, b_gate, W_dir,
                                                   out_sca, out_vec, N);
}